// Student_Model_88837103550992
// MI455X (gfx1250) — compile-verified
//
#include <hip/hip_runtime.h>
#include <hip/hip_bf16.h>

#define N_NODES 20000
#define FEAT_D 256
#define TWO_D 512
#define N_EDGES 320000
#define M_TILES (N_NODES / 16)        // 1250
#define MTILES_PER_BLOCK 80           // 8 waves x 10 tiles

typedef __attribute__((ext_vector_type(2))) float v2f;
typedef __attribute__((ext_vector_type(8))) float v8f;
typedef __attribute__((ext_vector_type(4))) unsigned int v4u;
typedef __attribute__((ext_vector_type(4))) int v4i;
typedef __attribute__((ext_vector_type(8))) int v8i;

// ---------------------------------------------------------------------------
// Workspace layout (floats):
//   [0, 20000)                    deg -> dinv (in place)
//   [20000, 20000+20000*256)      y = A_norm @ x   (20000 x 256)
//   [20000+20000*256, +512)       h = sum_nodes relu(y@W + b)   (512)
// ---------------------------------------------------------------------------
#define WS_DINV 0
#define WS_Y (N_NODES)
#define WS_H (N_NODES + (size_t)N_NODES * FEAT_D)
#define WS_TOTAL (WS_H + TWO_D)

__global__ void k_zero(float* __restrict__ ws, size_t n) {
  size_t i = (size_t)blockIdx.x * blockDim.x + threadIdx.x;
  size_t stride = (size_t)gridDim.x * blockDim.x;
  for (; i < n; i += stride) ws[i] = 0.0f;
}

__global__ void k_degree(const long long* __restrict__ edge, float* __restrict__ deg) {
  int e = blockIdx.x * blockDim.x + threadIdx.x;
  if (e < N_EDGES) {
    int dst = (int)edge[(size_t)e * 2 + 1];
    atomicAdd(&deg[dst], 1.0f);
  }
}

__global__ void k_dinv(float* __restrict__ deg) {
  int i = blockIdx.x * blockDim.x + threadIdx.x;
  if (i < N_NODES) {
    // self-loop contributes +1; deg >= 1 so the where()/max() guards are moot
    deg[i] = rsqrtf(deg[i] + 1.0f);
  }
}

// One block per (edge | self-loop); 256 threads = one feature channel each.
// y[dst] += dinv[src]*dinv[dst] * x[src]
__global__ void k_scatter(const float* __restrict__ x,
                          const long long* __restrict__ edge,
                          const float* __restrict__ dinv,
                          float* __restrict__ y) {
  int e = blockIdx.x;
  int c = threadIdx.x;
  int src, dst;
  if (e < N_EDGES) {
    src = (int)edge[(size_t)e * 2 + 0];
    dst = (int)edge[(size_t)e * 2 + 1];
  } else {
    src = dst = e - N_EDGES;  // self loop
  }
  float norm = dinv[src] * dinv[dst];
  atomicAdd(&y[(size_t)dst * FEAT_D + c], norm * x[(size_t)src * FEAT_D + c]);
}

// ---------------------------------------------------------------------------
// Block: fixed 16-column slab of W staged once into LDS by the Tensor Data
// Mover; 8 waves each sweep 10 M-tiles of 16 rows. Per tile: 64 x
// V_WMMA_F32_16X16X4_F32 with A from global (b64) and B from LDS (ds_load).
// Epilogue fuses bias + relu + row-reduction, atomicAdd into h[512].
//
// A (16x4 f32) lane layout: lanes 0-15 hold M rows with K={k0,k0+1};
// lanes 16-31 same rows with K={k0+2,k0+3}. B (4x16) mirrors across N.
// C/D: lane<16 col=lane rows 0..7 in v[0..7]; lane>=16 rows 8..15.
// ---------------------------------------------------------------------------
__global__ void __launch_bounds__(256)
k_gemm_relu_colsum(const float* __restrict__ y,
                   const float* __restrict__ W,   // (256, 512) row-major
                   const float* __restrict__ bias,
                   float* __restrict__ h) {
  __shared__ float ldsB[FEAT_D * 16];  // [k][n_local], 16 KB, 64B rows

  const int tid = threadIdx.x;
  const int wave = tid >> 5;
  const int lane = tid & 31;
  const int n0 = blockIdx.x * 16;

  // --- TDM: DMA W[0:256, n0:n0+16] (2D strided tile) into LDS -------------
  if (tid < 32) {  // one wave issues the descriptor (TDM ignores EXEC)
    unsigned lds_addr = (unsigned)(uintptr_t)(void*)&ldsB[0];
    unsigned long long ga = (unsigned long long)(uintptr_t)(const void*)(W + n0);

    v4u g0;
    g0[0] = 1u;                                        // count=1 (valid user D#)
    g0[1] = lds_addr;                                  // lds_addr [63:32]
    g0[2] = (unsigned)(ga & 0xffffffffull);            // global_addr [95:64]
    g0[3] = (unsigned)((ga >> 32) & 0x1ffffffull)      // global_addr [120:96]
            | (2u << 30);                              // type=2 ("image")

    v8i g1;
    g1[0] = (2 << 16);          // workgroup_mask=0 | data_size=2 (4 bytes)
    g1[1] = (int)(512u << 16);  // atomic_barrier_addr=0 | tensor_dim0 lo16 = 512
    g1[2] = (int)(256u << 16);  // tensor_dim0 hi16 = 0 | tensor_dim1 lo16 = 256
    g1[3] = (int)(16u << 16);   // tensor_dim1 hi16 = 0 | tile_dim0 = 16
    g1[4] = 256;                // tile_dim1 = 256 | tile_dim2 = 0
    g1[5] = 512;                // tensor_dim0_stride lo32 = 512 elements
    g1[6] = 0;                  // stride0 hi16 = 0 | tensor_dim1_stride lo16 = 0
    g1[7] = 0;                  // tensor_dim1_stride hi32 = 0

    v4i g2 = {0, 0, 0, 0};      // 2-D tensor: dims 2/3 unused (tile_dim3=0)
    v4i g3 = {0, 0, 0, 0};

#if defined(__clang_major__) && (__clang_major__ >= 23)
    v8i g4 = {0, 0, 0, 0, 0, 0, 0, 0};
    __builtin_amdgcn_tensor_load_to_lds(g0, g1, g2, g3, g4, 0);
#else
    __builtin_amdgcn_tensor_load_to_lds(g0, g1, g2, g3, 0);
#endif
    __builtin_amdgcn_s_wait_tensorcnt(0);
  }
  __syncthreads();

  const int half = lane >> 4;  // 0/1 -> K sub-pair
  const int l = lane & 15;
  const int n = n0 + l;        // B/C column handled by this lane
  const float bn = bias[n];

  for (int it = 0; it < MTILES_PER_BLOCK / 8; ++it) {
    const int mt = blockIdx.y * MTILES_PER_BLOCK + wave + 8 * it;
    if (mt >= M_TILES) break;               // wave-uniform
    const int m = mt * 16 + l;              // A row handled by this lane
    const float* __restrict__ yrow = y + (size_t)m * FEAT_D;
    // prefetch this lane's row for the next tile iteration (+128 rows)
    if (mt + 8 < M_TILES)
      __builtin_prefetch(yrow + (size_t)128 * FEAT_D, 0, 1);

    v8f c = {};
    for (int k0 = 0; k0 < FEAT_D; k0 += 4) {
      const int kb = k0 + 2 * half;
      v2f a = *(const v2f*)(yrow + kb);     // b64 global load
      v2f b;
      b.x = ldsB[kb * 16 + l];              // ds_load, conflict-free
      b.y = ldsB[(kb + 1) * 16 + l];
      c = __builtin_amdgcn_wmma_f32_16x16x4_f32(
          /*neg_a=*/false, a, /*neg_b=*/false, b,
          /*c_mod=*/(short)0, c, /*reuse_a=*/false, /*reuse_b=*/false);
    }

    float s = 0.0f;
#pragma unroll
    for (int r = 0; r < 8; ++r) {
      float v = c[r] + bn;
      s += v > 0.0f ? v : 0.0f;
    }
    atomicAdd(&h[n], s);  // lanes l and l+16 both hit column n; atomics combine
  }
}

// ---------------------------------------------------------------------------
// Tiny head network: convs on 32x16 image, softmaxes, MLP chain -> 2 outputs.
// Single block, 512 threads.
// ---------------------------------------------------------------------------
__device__ __forceinline__ void softmax512(const float* __restrict__ in,
                                           float* __restrict__ out,
                                           float* __restrict__ red) {
  int t = threadIdx.x;
  red[t] = in[t];
  __syncthreads();
  for (int s = 256; s > 0; s >>= 1) {
    if (t < s) red[t] = fmaxf(red[t], red[t + s]);
    __syncthreads();
  }
  float mx = red[0];
  __syncthreads();
  float e = __expf(in[t] - mx);
  red[t] = e;
  __syncthreads();
  for (int s = 256; s > 0; s >>= 1) {
    if (t < s) red[t] += red[t + s];
    __syncthreads();
  }
  float sum = red[0];
  __syncthreads();
  out[t] = e / sum;
}

__global__ void __launch_bounds__(512)
k_head(const float* __restrict__ h_in,
       const float* __restrict__ conv1_w, const float* __restrict__ conv1_b,
       const float* __restrict__ conv2_w, const float* __restrict__ conv2_b,
       const float* __restrict__ Wc1, const float* __restrict__ bc1,
       const float* __restrict__ Wc2, const float* __restrict__ bc2,
       const float* __restrict__ Wc3, const float* __restrict__ bc3,
       const float* __restrict__ Wp1, const float* __restrict__ bp1,
       const float* __restrict__ Wp2, const float* __restrict__ bp2,
       const float* __restrict__ Wp3, const float* __restrict__ bp3,
       const float* __restrict__ Wp4, const float* __restrict__ bp4,
       float* __restrict__ out) {
  __shared__ float h[TWO_D];
  __shared__ float bufA[TWO_D];
  __shared__ float bufB[TWO_D];
  __shared__ float comp[2 * TWO_D];
  __shared__ float red[TWO_D];

  const int t = threadIdx.x;  // 0..511
  h[t] = h_in[t];
  __syncthreads();

  // conv1 (3x3, SAME, 1 channel) over img[r][c] = h[r*16+c], 32x16
  {
    int r = t >> 4, cc = t & 15;
    float acc = conv1_b[0];
#pragma unroll
    for (int i = 0; i < 3; ++i)
#pragma unroll
      for (int j = 0; j < 3; ++j) {
        int rr = r + i - 1, c2 = cc + j - 1;
        if (rr >= 0 && rr < 32 && c2 >= 0 && c2 < 16)
          acc += conv1_w[i * 3 + j] * h[rr * 16 + c2];
      }
    bufA[t] = fmaxf(acc, 0.0f);
  }
  __syncthreads();

  // conv2
  {
    int r = t >> 4, cc = t & 15;
    float acc = conv2_b[0];
#pragma unroll
    for (int i = 0; i < 3; ++i)
#pragma unroll
      for (int j = 0; j < 3; ++j) {
        int rr = r + i - 1, c2 = cc + j - 1;
        if (rr >= 0 && rr < 32 && c2 >= 0 && c2 < 16)
          acc += conv2_w[i * 3 + j] * bufA[rr * 16 + c2];
      }
    bufB[t] = fmaxf(acc, 0.0f);
  }
  __syncthreads();

  // high_conf = softmax over all 512 conv outputs -> comp[0..511]
  softmax512(bufB, comp, red);
  __syncthreads();

  // f1 = relu(h @ Wc1 + bc1)
  {
    float acc = bc1[t];
    for (int k = 0; k < TWO_D; ++k) acc += h[k] * Wc1[(size_t)k * TWO_D + t];
    bufA[t] = fmaxf(acc, 0.0f);
  }
  __syncthreads();
  // f2 = relu(f1 @ Wc2 + bc2)
  {
    float acc = bc2[t];
    for (int k = 0; k < TWO_D; ++k) acc += bufA[k] * Wc2[(size_t)k * TWO_D + t];
    bufB[t] = fmaxf(acc, 0.0f);
  }
  __syncthreads();
  // front logits = f2 @ Wc3 + bc3 -> softmax -> comp[512..1023]
  {
    float acc = bc3[t];
    for (int k = 0; k < TWO_D; ++k) acc += bufB[k] * Wc3[(size_t)k * TWO_D + t];
    bufA[t] = acc;
  }
  __syncthreads();
  softmax512(bufA, comp + TWO_D, red);
  __syncthreads();

  // p1 = relu(complete(1024) @ Wp1 + bp1)   Wp1: (1024, 512)
  {
    float acc = bp1[t];
    for (int k = 0; k < 2 * TWO_D; ++k) acc += comp[k] * Wp1[(size_t)k * TWO_D + t];
    bufA[t] = fmaxf(acc, 0.0f);
  }
  __syncthreads();
  // p2 = relu(p1 @ Wp2 + bp2)   Wp2: (512, 256)
  if (t < FEAT_D) {
    float acc = bp2[t];
    for (int k = 0; k < TWO_D; ++k) acc += bufA[k] * Wp2[(size_t)k * FEAT_D + t];
    bufB[t] = fmaxf(acc, 0.0f);
  }
  __syncthreads();
  // p3 = relu(p2 @ Wp3 + bp3)   Wp3: (256, 128)   (reuse h as scratch)
  if (t < 128) {
    float acc = bp3[t];
    for (int k = 0; k < FEAT_D; ++k) acc += bufB[k] * Wp3[(size_t)k * 128 + t];
    h[t] = fmaxf(acc, 0.0f);
  }
  __syncthreads();
  // out = sigmoid(p3 @ Wp4 + bp4)   Wp4: (128, 2)
  if (t < 2) {
    float acc = bp4[t];
    for (int k = 0; k < 128; ++k) acc += h[k] * Wp4[(size_t)k * 2 + t];
    out[t] = 1.0f / (1.0f + __expf(-acc));
  }
}

extern "C" void kernel_launch(void* const* d_in, const int* in_sizes, int n_in,
                              void* d_out, int out_size, void* d_ws, size_t ws_size,
                              hipStream_t stream) {
  const float* x = (const float*)d_in[0];
  const long long* edge = (const long long*)d_in[1];
  const float* W_gcn = (const float*)d_in[2];
  const float* b_gcn = (const float*)d_in[3];
  const float* conv1_w = (const float*)d_in[4];
  const float* conv1_b = (const float*)d_in[5];
  const float* conv2_w = (const float*)d_in[6];
  const float* conv2_b = (const float*)d_in[7];
  const float* Wc1 = (const float*)d_in[8];
  const float* bc1 = (const float*)d_in[9];
  const float* Wc2 = (const float*)d_in[10];
  const float* bc2 = (const float*)d_in[11];
  const float* Wc3 = (const float*)d_in[12];
  const float* bc3 = (const float*)d_in[13];
  const float* Wp1 = (const float*)d_in[14];
  const float* bp1 = (const float*)d_in[15];
  const float* Wp2 = (const float*)d_in[16];
  const float* bp2 = (const float*)d_in[17];
  const float* Wp3 = (const float*)d_in[18];
  const float* bp3 = (const float*)d_in[19];
  const float* Wp4 = (const float*)d_in[20];
  const float* bp4 = (const float*)d_in[21];

  float* ws = (float*)d_ws;
  float* dinv = ws + WS_DINV;
  float* y = ws + WS_Y;
  float* h = ws + WS_H;
  float* out = (float*)d_out;

  (void)in_sizes; (void)n_in; (void)out_size; (void)ws_size;

  // 1) zero accumulators (deg, y, h)
  k_zero<<<4096, 256, 0, stream>>>(ws, WS_TOTAL);
  // 2) in-degree
  k_degree<<<(N_EDGES + 255) / 256, 256, 0, stream>>>(edge, dinv);
  // 3) dinv = rsqrt(deg + 1)
  k_dinv<<<(N_NODES + 255) / 256, 256, 0, stream>>>(dinv);
  // 4) y = A_norm @ x  (edges + self loops)
  k_scatter<<<N_EDGES + N_NODES, 256, 0, stream>>>(x, edge, dinv, y);
  // 5) h = sum_nodes relu(y @ W_gcn + b_gcn)  via TDM-staged f32 WMMA tiles
  k_gemm_relu_colsum<<<dim3(TWO_D / 16, (M_TILES + MTILES_PER_BLOCK - 1) / MTILES_PER_BLOCK),
                       256, 0, stream>>>(y, W_gcn, b_gcn, h);
  // 6) tiny head network -> 2 outputs
  k_head<<<1, 512, 0, stream>>>(h, conv1_w, conv1_b, conv2_w, conv2_b,
                                Wc1, bc1, Wc2, bc2, Wc3, bc3,
                                Wp1, bp1, Wp2, bp2, Wp3, bp3, Wp4, bp4, out);
}